// RGCN_84439057039591
// MI455X (gfx1250) — compile-verified
//
#include <hip/hip_runtime.h>
#include <hip/hip_bf16.h>

#define N_ENT    100000
#define N_REL    16
#define DIM      128
#define N_LAYERS 2
#define N_BASES  4
#define N_EDGES  640000
#define BATCH    1024
#define N_NEG    64
#define BN_EPS   1e-5f
#define ET       64     // edges per block in the message kernel

typedef __attribute__((ext_vector_type(2))) float v2f;
typedef __attribute__((ext_vector_type(8))) float v8f;

#define XS_STRIDE 132   // 128 + 4 pad -> bank (4*row + k) % 64, conflict-free

#define WMMA_F32(a, b, c) \
    __builtin_amdgcn_wmma_f32_16x16x4_f32(false, (a), false, (b), (short)0, (c), false, false)

// ---------------------------------------------------------------- utilities
__global__ void k_zero_f(float* __restrict__ p, int n) {
    int i = blockIdx.x * blockDim.x + threadIdx.x;
    if (i < n) p[i] = 0.0f;
}
__global__ void k_zero_i(int* __restrict__ p, int n) {
    int i = blockIdx.x * blockDim.x + threadIdx.x;
    if (i < n) p[i] = 0;
}

// ------------------------------------------------- degree + relation sort
__global__ void k_deg(const int* __restrict__ edge_index, float* __restrict__ deg) {
    int e = blockIdx.x * blockDim.x + threadIdx.x;
    if (e < N_EDGES) atomicAdd(&deg[edge_index[N_EDGES + e]], 1.0f);
}
__global__ void k_hist(const int* __restrict__ edge_type, int* __restrict__ cnt) {
    int e = blockIdx.x * blockDim.x + threadIdx.x;
    if (e < N_EDGES) atomicAdd(&cnt[edge_type[e]], 1);
}
__global__ void k_scan(const int* __restrict__ cnt, int* __restrict__ offs) {
    if (threadIdx.x == 0) {
        int a = 0;
        for (int r = 0; r < N_REL; ++r) { offs[r] = a; a += cnt[r]; }
        offs[N_REL] = a;
    }
}
__global__ void k_scatter(const int* __restrict__ edge_type, const int* __restrict__ offs,
                          int* __restrict__ cursor, int* __restrict__ sorted_eid) {
    int e = blockIdx.x * blockDim.x + threadIdx.x;
    if (e < N_EDGES) {
        int r = edge_type[e];
        int p = offs[r] + atomicAdd(&cursor[r], 1);
        sorted_eid[p] = e;
    }
}

// ------------------------------------------- W[r] = sum_b coeff[r,b]*basis[b]
__global__ void k_build_W(const float* __restrict__ bases, const float* __restrict__ coeffs,
                          float* __restrict__ W) {
    int idx = blockIdx.x * blockDim.x + threadIdx.x;
    if (idx >= N_REL * DIM * DIM) return;
    int r  = idx >> 14;       // /(128*128)
    int ij = idx & 16383;
    float s = 0.0f;
#pragma unroll
    for (int b = 0; b < N_BASES; ++b)
        s += coeffs[r * N_BASES + b] * bases[b * DIM * DIM + ij];
    W[idx] = s;
}

// --------- stage a 128x128 W matrix transposed into LDS: Wt[c*132 + k] = W[k*128 + c]
__device__ __forceinline__ void stage_Wt(const float* __restrict__ Wsrc, float* __restrict__ Wt,
                                         int t) {
#pragma unroll
    for (int i = 0; i < 16; ++i) {
        const int idx4 = t + i * 256;        // 0..4095 float4s
        const int k = idx4 >> 5;             // source row
        const int c = (idx4 & 31) * 4;       // source col
        const float4 w = *(const float4*)(Wsrc + (size_t)k * DIM + c);
        Wt[(c + 0) * XS_STRIDE + k] = w.x;
        Wt[(c + 1) * XS_STRIDE + k] = w.y;
        Wt[(c + 2) * XS_STRIDE + k] = w.z;
        Wt[(c + 3) * XS_STRIDE + k] = w.w;
    }
}

// ------------------------------------------------- self loop: acc = x@Wsl + b
// block = 256 (8 waves); one block = 32 rows; wave w -> N tile w, 2 M subtiles
// (B fragment loaded once per k-step from transposed LDS copy, reused 2x).
__global__ __launch_bounds__(256)
void k_selfloop(const float* __restrict__ x, const float* __restrict__ wsl,
                const float* __restrict__ bsl, float* __restrict__ acc) {
    __shared__ float Xs[32 * XS_STRIDE];
    __shared__ float Wt[DIM * XS_STRIDE];
    const int tile = blockIdx.x;               // N_ENT/32 = 3125 blocks, exact
    const int t = threadIdx.x;
    stage_Wt(wsl, Wt, t);
    {   // stage 32x128 tile of x (float4)
        const int row = t >> 3;                // 0..31
        const int c4  = t & 7;
        const float4* src = (const float4*)(x + (size_t)(tile * 32 + row) * DIM);
#pragma unroll
        for (int j = 0; j < 4; ++j) {
            const int f4 = c4 + j * 8;         // 0..31
            *(float4*)&Xs[row * XS_STRIDE + f4 * 4] = src[f4];
        }
    }
    __syncthreads();
    const int lane = t & 31;
    const int half = lane >> 4, m = lane & 15;
    const int n0 = (t >> 5) * 16;
    const int col = n0 + m;
    v8f c0 = {}, c1 = {};
#pragma unroll
    for (int k0 = 0; k0 < DIM; k0 += 4) {
        const int k = k0 + 2 * half;           // even -> 8B aligned
        const v2f b  = *(const v2f*)&Wt[col * XS_STRIDE + k];
        const v2f a0 = *(const v2f*)&Xs[m * XS_STRIDE + k];
        const v2f a1 = *(const v2f*)&Xs[(16 + m) * XS_STRIDE + k];
        c0 = WMMA_F32(a0, b, c0);
        c1 = WMMA_F32(a1, b, c1);
    }
    const float bias = bsl[col];
#pragma unroll
    for (int g = 0; g < 8; ++g) {
        const int r0 = tile * 32 + g + 8 * half;
        acc[(size_t)r0 * DIM + col]        = c0[g] + bias;
        acc[(size_t)(r0 + 16) * DIM + col] = c1[g] + bias;
    }
}

// --------------------------- per-edge messages: acc[dst] += x[src] @ W[rel]
// edges sorted by relation; 64-edge tiles, W[rel] staged transposed in LDS,
// B fragment reused across 4 M subtiles (4 WMMAs per B load).
__global__ __launch_bounds__(256)
void k_edge_msgs(const float* __restrict__ x, const float* __restrict__ W,
                 const int* __restrict__ sorted_eid, const int* __restrict__ edge_index,
                 const int* __restrict__ edge_type, float* __restrict__ acc) {
    __shared__ float Xs[ET * XS_STRIDE];
    __shared__ float Wt[DIM * XS_STRIDE];
    __shared__ int s_src[ET], s_dst[ET], s_rel[ET];
    __shared__ int s_segStart[ET], s_segEnd[ET], s_segRel[ET];
    __shared__ int s_nseg;
    const int tile = blockIdx.x;               // N_EDGES/ET = 10000 tiles, exact
    const int t = threadIdx.x;
    if (t < ET) {
        const int eid = sorted_eid[tile * ET + t];
        s_src[t] = edge_index[eid];
        s_dst[t] = edge_index[N_EDGES + eid];
        s_rel[t] = edge_type[eid];
    }
    __syncthreads();
    if (t == 0) {   // contiguous relation segments within the sorted tile
        int ns = 0, start = 0;
        for (int i = 1; i <= ET; ++i) {
            if (i == ET || s_rel[i] != s_rel[start]) {
                s_segRel[ns] = s_rel[start]; s_segStart[ns] = start; s_segEnd[ns] = i;
                ++ns; start = i;
            }
        }
        s_nseg = ns;
    }
    {   // gather 64 source rows of x into LDS (two phases of 32 rows)
        const int c4 = t & 7;
#pragma unroll
        for (int p = 0; p < 2; ++p) {
            const int row = p * 32 + (t >> 3);
            const float4* src = (const float4*)(x + (size_t)s_src[row] * DIM);
#pragma unroll
            for (int j = 0; j < 4; ++j) {
                const int f4 = c4 + j * 8;
                *(float4*)&Xs[row * XS_STRIDE + f4 * 4] = src[f4];
            }
        }
    }
    __syncthreads();
    const int lane = t & 31;
    const int half = lane >> 4, m = lane & 15;
    const int n0 = (t >> 5) * 16;
    const int col = n0 + m;
    const int nseg = s_nseg;                   // uniform across block
    for (int s = 0; s < nseg; ++s) {
        if (s > 0) __syncthreads();            // drain readers before restage
        stage_Wt(W + (size_t)s_segRel[s] * DIM * DIM, Wt, t);
        __syncthreads();
        v8f c0 = {}, c1 = {}, c2 = {}, c3 = {};
#pragma unroll
        for (int k0 = 0; k0 < DIM; k0 += 4) {
            const int k = k0 + 2 * half;
            const v2f b  = *(const v2f*)&Wt[col * XS_STRIDE + k];
            const v2f a0 = *(const v2f*)&Xs[m * XS_STRIDE + k];
            const v2f a1 = *(const v2f*)&Xs[(16 + m) * XS_STRIDE + k];
            const v2f a2 = *(const v2f*)&Xs[(32 + m) * XS_STRIDE + k];
            const v2f a3 = *(const v2f*)&Xs[(48 + m) * XS_STRIDE + k];
            c0 = WMMA_F32(a0, b, c0);
            c1 = WMMA_F32(a1, b, c1);
            c2 = WMMA_F32(a2, b, c2);
            c3 = WMMA_F32(a3, b, c3);
        }
        const int sA = s_segStart[s], sE = s_segEnd[s];
#pragma unroll
        for (int g = 0; g < 8; ++g) {
            const int rb = g + 8 * half;
            if (rb      >= sA && rb      < sE) atomicAdd(&acc[(size_t)s_dst[rb]      * DIM + col], c0[g]);
            if (rb + 16 >= sA && rb + 16 < sE) atomicAdd(&acc[(size_t)s_dst[rb + 16] * DIM + col], c1[g]);
            if (rb + 32 >= sA && rb + 32 < sE) atomicAdd(&acc[(size_t)s_dst[rb + 32] * DIM + col], c2[g]);
            if (rb + 48 >= sA && rb + 48 < sE) atomicAdd(&acc[(size_t)s_dst[rb + 48] * DIM + col], c3[g]);
        }
    }
}

// ------------------------------------------------------------ batch norm
__global__ __launch_bounds__(128)
void k_bn_stats(const float* __restrict__ acc, const float* __restrict__ deg,
                float* __restrict__ gsum, float* __restrict__ gsumsq) {
    const int d = threadIdx.x;                 // one dim per thread
    const int n0 = blockIdx.x * 500;           // 200 blocks * 500 nodes = 100000
    float s = 0.0f, ss = 0.0f;
    for (int i = 0; i < 500; ++i) {
        const int n = n0 + i;
        const float y = acc[(size_t)n * DIM + d] / fmaxf(deg[n], 1.0f);
        s += y; ss += y * y;
    }
    atomicAdd(&gsum[d], s);
    atomicAdd(&gsumsq[d], ss);
}

__global__ void k_bn_apply(const float* __restrict__ acc, const float* __restrict__ deg,
                           const float* __restrict__ gsum, const float* __restrict__ gsumsq,
                           const float* __restrict__ gamma, const float* __restrict__ beta,
                           float* __restrict__ xout) {
    const int idx = blockIdx.x * blockDim.x + threadIdx.x;
    if (idx >= N_ENT * DIM) return;
    const int d = idx & (DIM - 1);
    const int n = idx >> 7;
    const float invN = 1.0f / (float)N_ENT;
    const float mu  = gsum[d] * invN;
    const float var = gsumsq[d] * invN - mu * mu;
    const float y = acc[idx] / fmaxf(deg[n], 1.0f);
    const float v = (y - mu) * rsqrtf(var + BN_EPS) * gamma[d] + beta[d];
    xout[idx] = fmaxf(v, 0.0f);
}

// ------------------------------------------------------------ TransE scoring
__global__ __launch_bounds__(128)
void k_score(const float* __restrict__ ent, const float* __restrict__ relt,
             const int* __restrict__ head, const int* __restrict__ rel,
             const int* __restrict__ tail, const int* __restrict__ neg,
             float* __restrict__ pos_out, float* __restrict__ neg_out) {
    __shared__ float hr[DIM];
    __shared__ float sred[4];
    const int b = blockIdx.x;
    const int t = threadIdx.x;
    const float hv = ent[(size_t)head[b] * DIM + t] + relt[(size_t)rel[b] * DIM + t];
    hr[t] = hv;
    float v = hv - ent[(size_t)tail[b] * DIM + t];
    v = v * v;
#pragma unroll
    for (int off = 16; off > 0; off >>= 1) v += __shfl_down(v, off, 32);
    if ((t & 31) == 0) sred[t >> 5] = v;
    __syncthreads();
    if (t == 0) pos_out[b] = -sqrtf(sred[0] + sred[1] + sred[2] + sred[3]);
    __syncthreads();
    for (int k = 0; k < N_NEG; ++k) {
        const int ni = neg[b * N_NEG + k];
        float d = hr[t] - ent[(size_t)ni * DIM + t];
        float w = d * d;
#pragma unroll
        for (int off = 16; off > 0; off >>= 1) w += __shfl_down(w, off, 32);
        if ((t & 31) == 0) sred[t >> 5] = w;
        __syncthreads();
        if (t == 0) neg_out[b * N_NEG + k] = -sqrtf(sred[0] + sred[1] + sred[2] + sred[3]);
        __syncthreads();
    }
}

// ---------------------------------------------------------------- launcher
extern "C" void kernel_launch(void* const* d_in, const int* in_sizes, int n_in,
                              void* d_out, int out_size, void* d_ws, size_t ws_size,
                              hipStream_t stream) {
    const int*   head_idx      = (const int*)d_in[0];
    const int*   relation_idx  = (const int*)d_in[1];
    const int*   tail_idx      = (const int*)d_in[2];
    const int*   negative_idx  = (const int*)d_in[3];
    const int*   edge_index    = (const int*)d_in[4];
    const int*   edge_type     = (const int*)d_in[5];
    const float* entity_table  = (const float*)d_in[6];
    const float* relation_table= (const float*)d_in[7];
    const float* bases         = (const float*)d_in[8];
    const float* coeffs        = (const float*)d_in[9];
    const float* w_sl          = (const float*)d_in[10];
    const float* b_sl          = (const float*)d_in[11];
    const float* gamma         = (const float*)d_in[12];
    const float* beta          = (const float*)d_in[13];
    (void)in_sizes; (void)n_in; (void)out_size; (void)ws_size;

    char* ws = (char*)d_ws;
    size_t off = 0;
    auto take = [&](size_t bytes) -> char* {
        char* p = ws + off;
        off = (off + bytes + 255) & ~(size_t)255;
        return p;
    };
    float* W      = (float*)take((size_t)N_REL * DIM * DIM * 4);   //   1.0 MB
    float* deg    = (float*)take((size_t)N_ENT * 4);               //   0.4 MB
    int*   sorted = (int*)  take((size_t)N_EDGES * 4);             //   2.6 MB
    int*   cnt    = (int*)  take(64 * 4);                          // cnt|offs|cursor
    int*   offs   = cnt + 16;
    int*   cursor = cnt + 33;
    float* acc    = (float*)take((size_t)N_ENT * DIM * 4);         //  51.2 MB
    float* x1     = (float*)take((size_t)N_ENT * DIM * 4);         //  51.2 MB
    float* x2     = (float*)take((size_t)N_ENT * DIM * 4);         //  51.2 MB
    float* gsum   = (float*)take(DIM * 4);
    float* gsumsq = (float*)take(DIM * 4);

    // degree + relation counting sort (shared across layers)
    k_zero_f<<<(N_ENT + 255) / 256, 256, 0, stream>>>(deg, N_ENT);
    k_zero_i<<<1, 64, 0, stream>>>(cnt, 64);
    k_deg    <<<(N_EDGES + 255) / 256, 256, 0, stream>>>(edge_index, deg);
    k_hist   <<<(N_EDGES + 255) / 256, 256, 0, stream>>>(edge_type, cnt);
    k_scan   <<<1, 32, 0, stream>>>(cnt, offs);
    k_scatter<<<(N_EDGES + 255) / 256, 256, 0, stream>>>(edge_type, offs, cursor, sorted);

    for (int l = 0; l < N_LAYERS; ++l) {
        const float* xin  = (l == 0) ? entity_table : x1;
        float*       xout = (l == 0) ? x1 : x2;
        k_build_W<<<(N_REL * DIM * DIM + 255) / 256, 256, 0, stream>>>(
            bases + (size_t)l * N_BASES * DIM * DIM, coeffs + l * N_REL * N_BASES, W);
        k_selfloop<<<N_ENT / 32, 256, 0, stream>>>(
            xin, w_sl + (size_t)l * DIM * DIM, b_sl + l * DIM, acc);
        k_edge_msgs<<<N_EDGES / ET, 256, 0, stream>>>(
            xin, W, sorted, edge_index, edge_type, acc);
        k_zero_f<<<1, 128, 0, stream>>>(gsum, DIM);
        k_zero_f<<<1, 128, 0, stream>>>(gsumsq, DIM);
        k_bn_stats<<<200, 128, 0, stream>>>(acc, deg, gsum, gsumsq);
        k_bn_apply<<<(N_ENT * DIM + 255) / 256, 256, 0, stream>>>(
            acc, deg, gsum, gsumsq, gamma + l * DIM, beta + l * DIM, xout);
    }

    float* pos_out = (float*)d_out;
    float* neg_out = pos_out + BATCH;
    k_score<<<BATCH, DIM, 0, stream>>>(x2, relation_table, head_idx, relation_idx,
                                       tail_idx, negative_idx, pos_out, neg_out);
}